// HyperGsysHGNN_30142080484150
// MI455X (gfx1250) — compile-verified
//
#include <hip/hip_runtime.h>

typedef float v2f __attribute__((ext_vector_type(2)));
typedef float v8f __attribute__((ext_vector_type(8)));

#define N_VERT 100000
#define N_EDGE 50000
#define NNZ    3200000
#define IN_CH  256
#define OUT_CH 64
#define LDT    (IN_CH + 4)   // padded LDS row stride (floats) to avoid bank conflicts

// ---------------------------------------------------------------------------
// Dense projection: Xp[n][o] = sum_i X[n][i] * W[o][i]
// One block = 16 vertex rows x 64 out channels; 4 waves, wave w owns cols 16w..16w+15.
// Uses V_WMMA_F32_16X16X4_F32 (full fp32, matches reference precision).
// ---------------------------------------------------------------------------
__global__ __launch_bounds__(128) void hgnn_gemm_wmma(const float* __restrict__ X,
                                                      const float* __restrict__ W,
                                                      float* __restrict__ Xp) {
  extern __shared__ float smem[];
  float* Xs = smem;              // [16][LDT]
  float* Ws = smem + 16 * LDT;   // [64][LDT]

  const int tid  = threadIdx.x;      // 0..127
  const int lane = tid & 31;
  const int wave = tid >> 5;         // 0..3
  const int m0   = blockIdx.x * 16;  // vertex row base (100000 % 16 == 0)

  // Cooperative LDS staging, float4-coalesced.
  // X tile: 16 rows x 256 floats
  for (int r = tid >> 6; r < 16; r += 2) {
    const int c4 = tid & 63;
    float4 v = ((const float4*)(X + (size_t)(m0 + r) * IN_CH))[c4];
    float* dst = Xs + r * LDT + c4 * 4;
    dst[0] = v.x; dst[1] = v.y; dst[2] = v.z; dst[3] = v.w;
  }
  // Full W: 64 rows x 256 floats (64 KB, L2-resident source)
  for (int r = tid >> 6; r < 64; r += 2) {
    const int c4 = tid & 63;
    float4 v = ((const float4*)(W + (size_t)r * IN_CH))[c4];
    float* dst = Ws + r * LDT + c4 * 4;
    dst[0] = v.x; dst[1] = v.y; dst[2] = v.z; dst[3] = v.w;
  }
  __syncthreads();

  const int n0   = wave * 16;
  const int half = lane >> 4;   // 0: lanes 0-15 (K=k,k+1), 1: lanes 16-31 (K=k+2,k+3)
  const int mrow = lane & 15;   // A-matrix row supplied by this lane
  const int ncol = lane & 15;   // B/C/D column owned by this lane

  v8f acc = {};
  for (int k = 0; k < IN_CH; k += 4) {
    const int ka = k + half * 2;
    v2f a, b;
    // A 16x4 f32 layout (ISA 7.12.2): v0 = K even, v1 = K odd within lane-half
    a.x = Xs[mrow * LDT + ka];
    a.y = Xs[mrow * LDT + ka + 1];
    // B 4x16 f32: B[kb][n] = W[n0+n][kb]
    b.x = Ws[(n0 + ncol) * LDT + ka];
    b.y = Ws[(n0 + ncol) * LDT + ka + 1];
    acc = __builtin_amdgcn_wmma_f32_16x16x4_f32(false, a, false, b,
                                                (short)0, acc, false, false);
  }

  // C/D 16x16 f32 layout: VGPR r -> M = r + 8*half, N = ncol
  float* outBase = Xp + (size_t)m0 * OUT_CH + n0;
  #pragma unroll
  for (int r = 0; r < 8; ++r) {
    const int M = r + 8 * half;
    outBase[(size_t)M * OUT_CH + ncol] = acc[r];
  }
}

// ---------------------------------------------------------------------------
__global__ void zero_f32(float* __restrict__ p, int n) {
  int i = blockIdx.x * blockDim.x + threadIdx.x;
  if (i < n) p[i] = 0.0f;
}

// Stage 1: Xe[e] += Xp[v]   (16 lanes per nnz, float4 per lane -> coalesced gather)
__global__ void hgnn_scatter_v2e(const float* __restrict__ Xp,
                                 const int* __restrict__ v_idx,
                                 const int* __restrict__ e_idx,
                                 float* __restrict__ Xe) {
  const int gid  = blockIdx.x * blockDim.x + threadIdx.x;
  const int i    = gid >> 4;
  const int part = gid & 15;
  if (i >= NNZ) return;
  const int v = v_idx[i];
  const int e = e_idx[i];
  const float4 x = ((const float4*)(Xp + (size_t)v * OUT_CH))[part];
  float* dst = Xe + (size_t)e * OUT_CH + part * 4;
  atomicAdd(dst + 0, x.x);
  atomicAdd(dst + 1, x.y);
  atomicAdd(dst + 2, x.z);
  atomicAdd(dst + 3, x.w);
}

// Stage 2: Xv[v] += (degE[e]*Wdiag[e]) * Xe[e]   (edge scale folded into gather)
__global__ void hgnn_scatter_e2v(const float* __restrict__ Xe,
                                 const float* __restrict__ degE,
                                 const float* __restrict__ Wdiag,
                                 const int* __restrict__ v_idx,
                                 const int* __restrict__ e_idx,
                                 float* __restrict__ Xv) {
  const int gid  = blockIdx.x * blockDim.x + threadIdx.x;
  const int i    = gid >> 4;
  const int part = gid & 15;
  if (i >= NNZ) return;
  const int v = v_idx[i];
  const int e = e_idx[i];
  const float s = degE[e] * Wdiag[e];
  const float4 x = ((const float4*)(Xe + (size_t)e * OUT_CH))[part];
  float* dst = Xv + (size_t)v * OUT_CH + part * 4;
  atomicAdd(dst + 0, x.x * s);
  atomicAdd(dst + 1, x.y * s);
  atomicAdd(dst + 2, x.z * s);
  atomicAdd(dst + 3, x.w * s);
}

// Final per-vertex scale: Xv[v][c] *= degV[v]
__global__ void hgnn_scale_v(float* __restrict__ Xv, const float* __restrict__ degV) {
  const int i = blockIdx.x * blockDim.x + threadIdx.x;
  if (i < N_VERT * OUT_CH) Xv[i] *= degV[i >> 6];  // OUT_CH == 64
}

// ---------------------------------------------------------------------------
extern "C" void kernel_launch(void* const* d_in, const int* in_sizes, int n_in,
                              void* d_out, int out_size, void* d_ws, size_t ws_size,
                              hipStream_t stream) {
  const float* X     = (const float*)d_in[0];
  const float* W     = (const float*)d_in[1];
  const float* degE  = (const float*)d_in[2];
  const float* degV  = (const float*)d_in[3];
  const float* Wdiag = (const float*)d_in[4];
  const int*   v_idx = (const int*)d_in[5];
  const int*   e_idx = (const int*)d_in[6];
  float* out = (float*)d_out;

  float* Xp = (float*)d_ws;                         // 100000*64 f32 = 25.6 MB
  float* Xe = Xp + (size_t)N_VERT * OUT_CH;         //  50000*64 f32 = 12.8 MB

  const int nXe = N_EDGE * OUT_CH;   // 3.2M
  const int nXv = N_VERT * OUT_CH;   // 6.4M

  // Zero accumulators (deterministic, graph-safe)
  zero_f32<<<(nXe + 255) / 256, 256, 0, stream>>>(Xe, nXe);
  zero_f32<<<(nXv + 255) / 256, 256, 0, stream>>>(out, nXv);

  // Dense projection via WMMA f32
  const size_t shmemBytes = (size_t)(16 + 64) * LDT * sizeof(float);  // 83.2 KB
  hgnn_gemm_wmma<<<N_VERT / 16, 128, shmemBytes, stream>>>(X, W, Xp);

  // Scatter stages
  const int scatterThreads = NNZ * 16;
  hgnn_scatter_v2e<<<(scatterThreads + 255) / 256, 256, 0, stream>>>(Xp, v_idx, e_idx, Xe);
  hgnn_scatter_e2v<<<(scatterThreads + 255) / 256, 256, 0, stream>>>(Xe, degE, Wdiag, v_idx, e_idx, out);

  // Final degV scale
  hgnn_scale_v<<<(nXv + 255) / 256, 256, 0, stream>>>(out, degV);
}